// GraphConvGRU_87608742904452
// MI455X (gfx1250) — compile-verified
//
#include <hip/hip_runtime.h>

typedef __attribute__((ext_vector_type(2))) float v2f;
typedef __attribute__((ext_vector_type(8))) float v8f;

#define ROWS      66    /* 22 nodes * 3 hid */
#define A_STRIDE  68    /* padded K extent of fused operator */
#define H_STRIDE  20    /* LDS col stride of h tile (bank-safe) */
#define H_ROWS    68    /* padded K rows of h tile */
#define T_STEPS   100
#define OUT_ROW   6600  /* 100 * 66 */

__constant__ float DINV[22] = {
  0.27735009811f, 0.5f, 0.5f, 0.44721359550f, 0.5f, 0.5f, 0.44721359550f,
  0.5f, 0.5f, 0.27735009811f, 0.5f, 0.5f, 0.44721359550f, 0.5f, 0.5f,
  0.44721359550f, 0.5f, 0.5f, 0.5f, 0.5f, 0.5f, 0.5f };

__constant__ int CLQ[5][6] = {
  {0,2,5,8,11,-1},{0,1,4,7,10,-1},{0,3,6,9,12,15},
  {9,14,17,19,21,-1},{9,13,16,18,20,-1}};

__device__ __forceinline__ float fsig(float v){
  // 1/(1+e^-v) via v_exp_f32 (exp2) + v_rcp_f32
  return __builtin_amdgcn_rcpf(1.0f + __builtin_amdgcn_exp2f(-1.44269504089f * v));
}
__device__ __forceinline__ float ftanh(float v){
#if __has_builtin(__builtin_amdgcn_tanhf)
  return __builtin_amdgcn_tanhf(v);
#else
  return 1.0f - 2.0f * __builtin_amdgcn_rcpf(__builtin_amdgcn_exp2f(2.88539008178f * v) + 1.0f);
#endif
}

__global__ __launch_bounds__(128) void gcn_gru_fused(
    const float* __restrict__ x,
    const float* __restrict__ wrW, const float* __restrict__ wrb,
    const float* __restrict__ wzW, const float* __restrict__ wzb,
    const float* __restrict__ whW, const float* __restrict__ whb,
    const float* __restrict__ gW,  const float* __restrict__ gbias,
    float* __restrict__ out)
{
  __shared__ float    sA[80 * A_STRIDE];          // fused (A_HAT (x) gcn_W) operator, WMMA-A source
  __shared__ float    sH[4][H_ROWS * H_STRIDE];   // per-wave hidden-state tile (K x 16 cols)
  __shared__ float    sWm[3][96];
  __shared__ float    sBm[3][3];
  __shared__ float    sGB[3];
  __shared__ unsigned sAdj[22];

  const int tid = threadIdx.x;

  // ---- one-time block init: adjacency masks, weights, fused operator, zero h ----
  if (tid < 22){
    unsigned m = 0u;
    #pragma unroll
    for (int q = 0; q < 5; q++){
      bool inq = false;
      #pragma unroll
      for (int e = 0; e < 6; e++) inq |= (CLQ[q][e] == tid);
      if (inq){
        #pragma unroll
        for (int e = 0; e < 6; e++){ int v = CLQ[q][e]; if (v >= 0 && v != tid) m |= (1u << v); }
      }
    }
    sAdj[tid] = m;
  }
  for (int i = tid; i < 96; i += 128){ sWm[0][i] = wrW[i]; sWm[1][i] = wzW[i]; sWm[2][i] = whW[i]; }
  if (tid < 3){ sBm[0][tid] = wrb[tid]; sBm[1][tid] = wzb[tid]; sBm[2][tid] = whb[tid]; sGB[tid] = gbias[tid]; }
  __syncthreads();

  for (int idx = tid; idx < 80 * A_STRIDE; idx += 128){
    int r = idx / A_STRIDE, c = idx - A_STRIDE * r;
    float v = 0.f;
    if (r < ROWS && c < ROWS){
      int n = r / 3, j = r - 3 * n, mm = c / 3, i = c - 3 * mm;
      if ((sAdj[n] >> mm) & 1u) v = DINV[n] * DINV[mm] * gW[i * 3 + j];
    }
    sA[idx] = v;  // zero padding rows/cols -> K-pad contributes 0, no NaN leakage
  }
  for (int idx = tid; idx < 4 * H_ROWS * H_STRIDE; idx += 128) (&sH[0][0])[idx] = 0.f;
  __syncthreads();

  const int  lane = tid & 31;
  const int  wv   = tid >> 5;
  const bool hi   = lane >= 16;
  const int  col  = lane & 15;          // batch column within this wave's 16-col tile
  const long bidx = (long)blockIdx.x * 64 + wv * 16 + col;

  // ---- one-time input projections for this lane's batch row ----
  float xr[3], xz[3], xh[3], gb[3];
  #pragma unroll
  for (int j = 0; j < 3; j++){ xr[j] = sBm[0][j]; xz[j] = sBm[1][j]; xh[j] = sBm[2][j]; gb[j] = sGB[j]; }
  {
    const float* xb = x + bidx * 32;
    #pragma unroll
    for (int k4 = 0; k4 < 8; k4++){
      float4 xv = ((const float4*)xb)[k4];
      float xs[4] = {xv.x, xv.y, xv.z, xv.w};
      #pragma unroll
      for (int q = 0; q < 4; q++){
        int k = k4 * 4 + q;
        #pragma unroll
        for (int j = 0; j < 3; j++){
          xr[j] = __builtin_fmaf(xs[q], sWm[0][k * 3 + j], xr[j]);
          xz[j] = __builtin_fmaf(xs[q], sWm[1][k * 3 + j], xz[j]);
          xh[j] = __builtin_fmaf(xs[q], sWm[2][k * 3 + j], xh[j]);
        }
      }
    }
  }

  float* myH = sH[wv];
  float h[5][8];
  #pragma unroll
  for (int mt = 0; mt < 5; mt++)
    #pragma unroll
    for (int r = 0; r < 8; r++) h[mt][r] = 0.f;

  const int koff = hi ? 2 : 0;   // f32 WMMA A/B: lane-half selects K+2
  const int rofs = hi ? 8 : 0;   // f32 WMMA D: lane-half selects M+8
  float* outp = out + (size_t)bidx * OUT_ROW;

  for (int t = 0; t < T_STEPS; t++){
    asm volatile("" ::: "memory");  // keep invariant A-operand LDS loads inside the loop (no LICM reg blowup)

    v8f acc[5];
    #pragma unroll
    for (int mt = 0; mt < 5; mt++) acc[mt] = (v8f){0.f,0.f,0.f,0.f,0.f,0.f,0.f,0.f};

    // gh(80x16) = M66(80x68) @ h(68x16) : 17 K-steps x 5 M-tiles of v_wmma_f32_16x16x4_f32
    #pragma unroll
    for (int kt = 0; kt < 17; kt++){
      const int kr = kt * 4 + koff;
      v2f bb;
      bb.x = myH[ kr      * H_STRIDE + col];
      bb.y = myH[(kr + 1) * H_STRIDE + col];
      #pragma unroll
      for (int mt = 0; mt < 5; mt++){
        const v2f a = *(const v2f*)&sA[(mt * 16 + col) * A_STRIDE + kr];
        acc[mt] = __builtin_amdgcn_wmma_f32_16x16x4_f32(
            false, a, false, bb, (short)0, acc[mt], false, false);
      }
    }

    // ---- GRU gating, elementwise in D layout ----
    #pragma unroll
    for (int mt = 0; mt < 5; mt++){
      #pragma unroll
      for (int r = 0; r < 8; r++){
        const int j0 = (mt * 16 + r) % 3;
        const int j1 = (j0 + 2) % 3;        // row%3 for the hi half
        const float bsel  = hi ? gb[j1] : gb[j0];
        const float xrsel = hi ? xr[j1] : xr[j0];
        const float xzsel = hi ? xz[j1] : xz[j0];
        const float xhsel = hi ? xh[j1] : xh[j0];
        const float gh = acc[mt][r] + bsel;
        const float rg = fsig(xrsel + gh);
        const float zg = fsig(xzsel + gh);
        const float ht = ftanh(__builtin_fmaf(rg, gh, xhsel));
        h[mt][r] = __builtin_fmaf(zg, ht - h[mt][r], h[mt][r]);
      }
    }

    // ---- re-materialize h in B layout via per-wave LDS tile (rows >= 66 stay zero) ----
    #pragma unroll
    for (int mt = 0; mt < 5; mt++){
      #pragma unroll
      for (int r = 0; r < 8; r++){
        const int row = mt * 16 + rofs + r;
        if (mt < 4)                 myH[row * H_STRIDE + col] = h[mt][r];
        else if (!hi && r < 2)      myH[row * H_STRIDE + col] = h[mt][r];
      }
    }

    // ---- stream this step's hidden state to out[b, t*66 + row], 8B-aligned b64 stores ----
    const size_t tb = (size_t)t * 66;
    #pragma unroll
    for (int mt = 0; mt < 4; mt++){
      #pragma unroll
      for (int r2 = 0; r2 < 4; r2++){
        float2 p; p.x = h[mt][2 * r2]; p.y = h[mt][2 * r2 + 1];
        *(float2*)(outp + tb + mt * 16 + rofs + 2 * r2) = p;
      }
    }
    if (!hi){
      float2 p; p.x = h[4][0]; p.y = h[4][1];
      *(float2*)(outp + tb + 64) = p;
    }
  }
}

extern "C" void kernel_launch(void* const* d_in, const int* in_sizes, int n_in,
                              void* d_out, int out_size, void* d_ws, size_t ws_size,
                              hipStream_t stream)
{
  const float* x   = (const float*)d_in[0];
  const float* wrW = (const float*)d_in[1];
  const float* wrb = (const float*)d_in[2];
  const float* wzW = (const float*)d_in[3];
  const float* wzb = (const float*)d_in[4];
  const float* whW = (const float*)d_in[5];
  const float* whb = (const float*)d_in[6];
  const float* gW  = (const float*)d_in[7];
  const float* gb  = (const float*)d_in[8];
  float* out = (float*)d_out;

  const int Btot = in_sizes[0] / 32;   // 16384
  const int nblk = Btot / 64;          // 64 batch columns per 128-thread block (4 waves x 16)

  gcn_gru_fused<<<dim3(nblk), dim3(128), 0, stream>>>(
      x, wrW, wrb, wzW, wzb, whW, whb, gW, gb, out);

  (void)n_in; (void)out_size; (void)d_ws; (void)ws_size;
}